// Transform_SVD_17849884082566
// MI455X (gfx1250) — compile-verified
//
#include <hip/hip_runtime.h>

// Transform_SVD: per-voxel 3x3 polar factor of the Jacobian of (identity + flow).
//
// Math: J = I + grad(flow) (jnp.gradient semantics). U@Vh from svd(J) is the
// unique orthogonal polar factor of J -> determinant-scaled Newton iteration
// X <- 0.5*(mu*X + C/(mu*det)) (C = cofactor matrix, X^-T = C/det), quadratic
// convergence, 7 iterations to f32 accuracy. Singular values for the
// distinctness mask via closed-form Cardano eigen-solve of J^T J.
//
// MI455X shape: flow (19 MB) fits in 192 MB L2 -> all 18 neighbor loads are
// cache hits; output (64 MB) is write-once -> non-temporal streaming stores.
// Kernel is VALU-bound (~800 f32 ops/voxel vs 52 B/voxel HBM), so: branch-free
// clamped-index gradients, native v_rcp/v_log/v_exp/v_cos transcendentals in
// the self-correcting early Newton steps, IEEE division only where the final
// answer depends on it.

namespace {
constexpr int Dd = 112, Hh = 128, Ww = 112;
constexpr int HW  = Hh * Ww;          // 14336
constexpr int DHW = Dd * HW;          // 1,605,632
}

__global__ __launch_bounds__(256) void polar_jac_kernel(
    const float* __restrict__ flow, float* __restrict__ out) {
  const int w = (blockIdx.x << 4) + threadIdx.x;   // 0..111
  const int h = (blockIdx.y << 4) + threadIdx.y;   // 0..127
  const int d = blockIdx.z;                        // 0..111
  const int n = (d * Hh + h) * Ww + w;

  // Branch-free gradient endpoints + scale (1 at edges, 1/2 interior).
  const int dm = max(d - 1, 0), dp = min(d + 1, Dd - 1);
  const int hm = max(h - 1, 0), hp = min(h + 1, Hh - 1);
  const int wm = max(w - 1, 0), wp = min(w + 1, Ww - 1);
  const float sd = (dp - dm == 2) ? 0.5f : 1.0f;
  const float sh = (hp - hm == 2) ? 0.5f : 1.0f;
  const float sw = (wp - wm == 2) ? 0.5f : 1.0f;

  const int n_hw   = h * Ww + w;
  const int off_dm = dm * HW + n_hw;
  const int off_dp = dp * HW + n_hw;
  const int off_hm = d * HW + hm * Ww + w;
  const int off_hp = d * HW + hp * Ww + w;
  const int off_wm = d * HW + h * Ww + wm;
  const int off_wp = d * HW + h * Ww + wp;

  // ---- J[c][a] = delta_ca + d(flow_c)/d(axis_a) ----
  float X[9];
#pragma unroll
  for (int c = 0; c < 3; ++c) {
    const float* f = flow + c * DHW;
    X[c * 3 + 0] = (f[off_dp] - f[off_dm]) * sd + (c == 0 ? 1.0f : 0.0f);
    X[c * 3 + 1] = (f[off_hp] - f[off_hm]) * sh + (c == 1 ? 1.0f : 0.0f);
    X[c * 3 + 2] = (f[off_wp] - f[off_wm]) * sw + (c == 2 ? 1.0f : 0.0f);
  }

  // ---- singular values via closed-form eigenvalues of A = J^T J ----
  const float a00 = X[0]*X[0] + X[3]*X[3] + X[6]*X[6];
  const float a11 = X[1]*X[1] + X[4]*X[4] + X[7]*X[7];
  const float a22 = X[2]*X[2] + X[5]*X[5] + X[8]*X[8];
  const float a01 = X[0]*X[1] + X[3]*X[4] + X[6]*X[7];
  const float a02 = X[0]*X[2] + X[3]*X[5] + X[6]*X[8];
  const float a12 = X[1]*X[2] + X[4]*X[5] + X[7]*X[8];

  const float q   = (a00 + a11 + a22) * (1.0f / 3.0f);
  const float p1  = a01*a01 + a02*a02 + a12*a12;
  const float b00 = a00 - q, b11 = a11 - q, b22 = a22 - q;
  const float p2  = b00*b00 + b11*b11 + b22*b22 + 2.0f * p1;
  const float p   = sqrtf(p2 * (1.0f / 6.0f));

  float maskv;
  if (p < 1e-18f) {
    maskv = 0.0f;  // A ~ q*I: all singular values equal -> dropped
  } else {
    const float ip   = __builtin_amdgcn_rcpf(p);   // mask path: 1-ulp rcp is plenty
    const float detM = b00 * (b11 * b22 - a12 * a12)
                     - a01 * (a01 * b22 - a12 * a02)
                     + a02 * (a01 * a12 - b11 * a02);
    float r = 0.5f * detM * ip * ip * ip;
    r = fminf(1.0f, fmaxf(-1.0f, r));
    const float phi  = acosf(r) * (1.0f / 3.0f);
    const float lam0 = q + 2.0f * p * __cosf(phi);                          // native v_cos
    const float lam2 = q + 2.0f * p * __cosf(phi + 2.0943951023931953f);    // +2pi/3
    const float lam1 = 3.0f * q - lam0 - lam2;
    const float s0 = sqrtf(fmaxf(lam0, 0.0f));
    const float s1 = sqrtf(fmaxf(lam1, 0.0f));
    const float s2 = sqrtf(fmaxf(lam2, 0.0f));
    maskv = ((s0 != s1) && (s0 != s2) && (s1 != s2)) ? 1.0f : 0.0f;
  }

  // ---- determinant-scaled Newton polar iteration: X -> orthogonal factor ----
#pragma unroll
  for (int it = 0; it < 7; ++it) {
    const float C0 = X[4]*X[8] - X[5]*X[7];
    const float C1 = X[5]*X[6] - X[3]*X[8];
    const float C2 = X[3]*X[7] - X[4]*X[6];
    const float C3 = X[2]*X[7] - X[1]*X[8];
    const float C4 = X[0]*X[8] - X[2]*X[6];
    const float C5 = X[1]*X[6] - X[0]*X[7];
    const float C6 = X[1]*X[5] - X[2]*X[4];
    const float C7 = X[2]*X[3] - X[0]*X[5];
    const float C8 = X[0]*X[4] - X[1]*X[3];
    float det = X[0]*C0 + X[1]*C1 + X[2]*C2;
    if (fabsf(det) < 1e-30f) det = (det < 0.0f) ? -1e-30f : 1e-30f;

    float ca, cb;
    if (it < 4) {
      // mu = |det|^(-1/3) determinantal scaling; early steps are
      // self-correcting, so raw v_log_f32/v_exp_f32/v_rcp_f32 is sufficient.
      const float mu = __builtin_amdgcn_exp2f(
          -(1.0f / 3.0f) * __builtin_amdgcn_logf(fabsf(det)));
      ca = 0.5f * mu;
      cb = 0.5f * mu * mu * __builtin_amdgcn_rcpf(det * mu * mu * mu);
    } else if (it < 5) {
      ca = 0.5f;
      cb = 0.5f * __builtin_amdgcn_rcpf(det);
    } else {
      // last two steps: IEEE division, full f32 accuracy in the result
      ca = 0.5f;
      cb = 0.5f / det;
    }

    X[0] = ca*X[0] + cb*C0;  X[1] = ca*X[1] + cb*C1;  X[2] = ca*X[2] + cb*C2;
    X[3] = ca*X[3] + cb*C3;  X[4] = ca*X[4] + cb*C4;  X[5] = ca*X[5] + cb*C5;
    X[6] = ca*X[6] + cb*C6;  X[7] = ca*X[7] + cb*C7;  X[8] = ca*X[8] + cb*C8;
  }

  // ---- R = (U Vh)^T = X^T ; stream out with non-temporal stores ----
  float* o = out + (size_t)n * 9u;
  __builtin_nontemporal_store(X[0], o + 0);
  __builtin_nontemporal_store(X[3], o + 1);
  __builtin_nontemporal_store(X[6], o + 2);
  __builtin_nontemporal_store(X[1], o + 3);
  __builtin_nontemporal_store(X[4], o + 4);
  __builtin_nontemporal_store(X[7], o + 5);
  __builtin_nontemporal_store(X[2], o + 6);
  __builtin_nontemporal_store(X[5], o + 7);
  __builtin_nontemporal_store(X[8], o + 8);
  __builtin_nontemporal_store(maskv, out + (size_t)DHW * 9u + n);
}

extern "C" void kernel_launch(void* const* d_in, const int* in_sizes, int n_in,
                              void* d_out, int out_size, void* d_ws, size_t ws_size,
                              hipStream_t stream) {
  (void)in_sizes; (void)n_in; (void)out_size; (void)d_ws; (void)ws_size;
  const float* flow = (const float*)d_in[0];   // (1,3,112,128,112) f32
  float* out = (float*)d_out;                  // N*9 floats (R) then N floats (mask)

  dim3 block(16, 16, 1);                       // 256 threads = 8 wave32
  dim3 grid(Ww / 16, Hh / 16, Dd);             // (7, 8, 112) — exact tiling
  polar_jac_kernel<<<grid, block, 0, stream>>>(flow, out);
}